// RelationTransform_79061757984918
// MI455X (gfx1250) — compile-verified
//
#include <hip/hip_runtime.h>
#include <hip/hip_bf16.h>

typedef __attribute__((ext_vector_type(2))) float v2f;
typedef __attribute__((ext_vector_type(4))) float v4f;
typedef __attribute__((ext_vector_type(8))) float v8f;

#define N_REL 16
#define EMB 512
#define LDS_STRIDE 516   // 512 + 4 pad: keeps float4 LDS stores 16B-aligned and
                         // spreads the 16 A-rows across distinct LDS banks (4*m % 64)

// d_ws int32 layout:
//  [0..15]   cnt[r]       histogram
//  [16..31]  off[r]       exclusive token prefix
//  [32..47]  cursor[r]    scatter cursors
//  [48..64]  tb[r] (17)   exclusive tile prefix; tb[16] = total tiles
//  [80.. )   idx[N]       token indices grouped by relation
#define WS_CNT 0
#define WS_OFF 16
#define WS_CUR 32
#define WS_TB  48
#define WS_IDX 80

__global__ void rt_init(int* ws) {
    int t = threadIdx.x;
    if (t < WS_IDX) ws[t] = 0;
}

__global__ void rt_hist(const int* __restrict__ rel, int* __restrict__ ws, int n) {
    int i = blockIdx.x * blockDim.x + threadIdx.x;
    if (i < n) atomicAdd(&ws[WS_CNT + rel[i]], 1);
}

__global__ void rt_scan(int* ws) {
    if (threadIdx.x == 0) {
        int run = 0, tiles = 0;
        for (int r = 0; r < N_REL; ++r) {
            ws[WS_OFF + r] = run;
            ws[WS_TB + r]  = tiles;
            int c = ws[WS_CNT + r];
            run   += c;
            tiles += (c + 15) >> 4;
        }
        ws[WS_TB + N_REL] = tiles;
    }
}

__global__ void rt_scatter(const int* __restrict__ rel, int* __restrict__ ws, int n) {
    int i = blockIdx.x * blockDim.x + threadIdx.x;
    if (i < n) {
        int r = rel[i];
        int p = atomicAdd(&ws[WS_CUR + r], 1);
        ws[WS_IDX + ws[WS_OFF + r] + p] = i;
    }
}

// Grid: (maxTiles, 4). Block: 128 threads = 4 waves.
// Each WG: one 16-token tile of one relation bucket x 128 output columns.
// Each wave: 16x32 output tile, K=512 via V_WMMA_F32_16X16X4_F32.
__global__ void __launch_bounds__(128)
rt_gemm(const float* __restrict__ X, const float* __restrict__ W,
        const int* __restrict__ ws, float* __restrict__ out) {
    __shared__ float As[16 * LDS_STRIDE];
    __shared__ int  s_tok[16];

    const int* tb = ws + WS_TB;
    int tile = blockIdx.x;
    if (tile >= tb[N_REL]) return;          // uniform per-WG: safe before barrier

    // locate relation bucket: tb[] is nondecreasing, tb[r] <= tile < tb[r+1]
    int r = 0;
    #pragma unroll
    for (int i = 1; i < N_REL; ++i) r += (tile >= tb[i]) ? 1 : 0;

    int tloc  = tile - tb[r];
    int cnt_r = ws[WS_CNT + r];
    int base  = ws[WS_OFF + r] + 16 * tloc;
    int v     = cnt_r - 16 * tloc;          // valid tokens in this tile (>=1)
    if (v > 16) v = 16;

    if (threadIdx.x < 16) {
        int j  = threadIdx.x;
        int jj = (j < v) ? j : (v - 1);     // clamp: duplicate loads, stores masked
        s_tok[j] = ws[WS_IDX + base + jj];
    }
    __syncthreads();

    // Stage the 16x512 fp32 A tile into LDS (gathered rows), float4 granularity.
    for (int q = threadIdx.x; q < 16 * 128; q += 128) {
        int row = q >> 7, c4 = q & 127;
        v4f val = *(const v4f*)(X + (size_t)s_tok[row] * EMB + 4 * c4);
        *(v4f*)(&As[row * LDS_STRIDE + 4 * c4]) = val;
    }
    __syncthreads();

    int lane  = threadIdx.x & 31;
    int wavei = threadIdx.x >> 5;
    int i16   = lane & 15;                  // M (A) / N (B,D) index of this lane
    int khalf = lane >> 4;                  // ISA: lanes 16-31 carry K+2..K+3
    int koff  = 2 * khalf;

    int n0 = blockIdx.y * 128 + wavei * 32; // this wave's 32-column slab
    const float* Wr  = W + (size_t)r * EMB * EMB;
    const float* b0p = Wr + (size_t)(n0 + i16) * EMB + koff;   // B[k][n] = Wr[n][k]
    const float* b1p = b0p + (size_t)16 * EMB;
    const float* ap  = &As[i16 * LDS_STRIDE + koff];

    v8f acc0 = {};
    v8f acc1 = {};
    #pragma unroll 4
    for (int kb = 0; kb < EMB; kb += 4) {
        v2f a  = *(const v2f*)(ap  + kb);   // ds_load_b64, bank-conflict free
        v2f b0 = *(const v2f*)(b0p + kb);   // L2-resident W slab
        v2f b1 = *(const v2f*)(b1p + kb);
        acc0 = __builtin_amdgcn_wmma_f32_16x16x4_f32(false, a, false, b0,
                                                     (short)0, acc0, false, false);
        acc1 = __builtin_amdgcn_wmma_f32_16x16x4_f32(false, a, false, b1,
                                                     (short)0, acc1, false, false);
    }

    // D layout: lane holds column i16, VGPR q -> token row q + 8*khalf.
    int n = n0 + i16;
    int mbase = 8 * khalf;
    #pragma unroll
    for (int q = 0; q < 8; ++q) {
        int m = mbase + q;
        if (m < v) {
            size_t o = (size_t)s_tok[m] * EMB + n;
            out[o]      = acc0[q];
            out[o + 16] = acc1[q];
        }
    }
}

extern "C" void kernel_launch(void* const* d_in, const int* in_sizes, int n_in,
                              void* d_out, int out_size, void* d_ws, size_t ws_size,
                              hipStream_t stream) {
    const float* X   = (const float*)d_in[0];   // node_emb [N,512] fp32
    const int*   rel = (const int*)  d_in[1];   // relation [N] int32
    const float* W   = (const float*)d_in[2];   // W [16,512,512] fp32
    float* out = (float*)d_out;
    int*   ws  = (int*)d_ws;

    int n = in_sizes[1];                        // 32768 tokens

    rt_init<<<1, 128, 0, stream>>>(ws);
    int blocks = (n + 255) / 256;
    rt_hist<<<blocks, 256, 0, stream>>>(rel, ws, n);
    rt_scan<<<1, 32, 0, stream>>>(ws);
    rt_scatter<<<blocks, 256, 0, stream>>>(rel, ws, n);

    int maxTiles = n / 16 + N_REL;              // upper bound on sum ceil(cnt_r/16)
    dim3 grid(maxTiles, EMB / 128);
    rt_gemm<<<grid, 128, 0, stream>>>(X, W, ws, out);
}